// AttentionRefineModuleNew_88175678587418
// MI455X (gfx1250) — compile-verified
//
#include <hip/hip_runtime.h>

// ---------------------------------------------------------------------------
// Shapes (fixed by the reference): B=32, C=2048, H=W=24 -> N=576, MID=256
// ---------------------------------------------------------------------------
constexpr int Bsz  = 32;
constexpr int Cch  = 2048;
constexpr int HW   = 576;    // N = H*W
constexpr int MIDc = 256;

typedef float v2f __attribute__((ext_vector_type(2)));
typedef float v8f __attribute__((ext_vector_type(8)));

// D = A(16x4 f32) * B(4x16 f32) + C(16x16 f32)   [CDNA5 V_WMMA_F32_16X16X4_F32]
#define WMMA_F32_16x16x4(a, b, c) \
    __builtin_amdgcn_wmma_f32_16x16x4_f32(false, (a), false, (b), (short)0, (c), false, false)

// ---------------------------------------------------------------------------
// Kernel 1: f1 = W1 x feat[b], f2 = W2 x feat[b]   (1x1 conv as channel GEMM)
//   M = MID(256), K = C(2048), N = HW(576).
//   Register blocking: one wave computes a 32x32 tile of BOTH f1 and f2
//   (2 M-tiles x 2 N-tiles x 2 gemms = 8 accumulators). The feat B-fragments
//   are shared by 4 WMMAs each; W fragments by 2. 8 WMMAs per 8 loads.
// ---------------------------------------------------------------------------
__global__ __launch_bounds__(32)
void k_conv1x1_pair(const float* __restrict__ feat,
                    const float* __restrict__ W1,
                    const float* __restrict__ W2,
                    float* __restrict__ f1,
                    float* __restrict__ f2)
{
    const int lane = threadIdx.x & 31;
    const int half = lane >> 4;        // 0: lanes 0-15, 1: lanes 16-31
    const int l    = lane & 15;
    const int nBase = blockIdx.x * 32; // column block over N
    const int mBase = blockIdx.y * 32; // row block over MID
    const int b     = blockIdx.z;

    const float* __restrict__ w1row0 = W1 + (size_t)(mBase + l) * Cch;
    const float* __restrict__ w1row1 = w1row0 + (size_t)16 * Cch;
    const float* __restrict__ w2row0 = W2 + (size_t)(mBase + l) * Cch;
    const float* __restrict__ w2row1 = w2row0 + (size_t)16 * Cch;
    const float* __restrict__ fcol   = feat + (size_t)b * Cch * HW + nBase + l;

    v8f c1[2][2] = {};
    v8f c2[2][2] = {};

#pragma unroll 2
    for (int k0 = 0; k0 < Cch; k0 += 4) {
        const int kk = k0 + 2 * half;
        // A fragments (W rows, contiguous over K -> b64 loads)
        v2f a1[2], a2[2];
        a1[0] = *(const v2f*)(w1row0 + kk);
        a1[1] = *(const v2f*)(w1row1 + kk);
        a2[0] = *(const v2f*)(w2row0 + kk);
        a2[1] = *(const v2f*)(w2row1 + kk);
        // B fragments (feat rows kk, kk+1; two 16-wide column tiles)
        v2f bf[2];
        bf[0].x = fcol[(size_t)kk * HW];
        bf[0].y = fcol[(size_t)(kk + 1) * HW];
        bf[1].x = fcol[(size_t)kk * HW + 16];
        bf[1].y = fcol[(size_t)(kk + 1) * HW + 16];
#pragma unroll
        for (int mi = 0; mi < 2; ++mi)
#pragma unroll
            for (int ni = 0; ni < 2; ++ni) {
                c1[mi][ni] = WMMA_F32_16x16x4(a1[mi], bf[ni], c1[mi][ni]);
                c2[mi][ni] = WMMA_F32_16x16x4(a2[mi], bf[ni], c2[mi][ni]);
            }
    }

    float* __restrict__ o1 = f1 + ((size_t)b * MIDc + mBase + 8 * half) * HW + nBase + l;
    float* __restrict__ o2 = f2 + ((size_t)b * MIDc + mBase + 8 * half) * HW + nBase + l;
#pragma unroll
    for (int mi = 0; mi < 2; ++mi)
#pragma unroll
        for (int ni = 0; ni < 2; ++ni)
#pragma unroll
            for (int r = 0; r < 8; ++r) {
                const size_t off = (size_t)(16 * mi + r) * HW + 16 * ni;
                o1[off] = c1[mi][ni][r];
                o2[off] = c2[mi][ni][r];
            }
}

// ---------------------------------------------------------------------------
// Kernel 2: attn[b] = f1[b]^T x f2[b]   (N x MID) * (MID x N) -> N x N
//   attn[i][j] = sum_m f1[m][i] * f2[m][j];  K = MID(256)
//   2x2 register blocking: 32x32 output per wave, fragments reused 2x.
// ---------------------------------------------------------------------------
__global__ __launch_bounds__(32)
void k_attn_scores(const float* __restrict__ f1,
                   const float* __restrict__ f2,
                   float* __restrict__ attn)
{
    const int lane = threadIdx.x & 31;
    const int half = lane >> 4;
    const int l    = lane & 15;
    const int jBase = blockIdx.x * 32;
    const int iBase = blockIdx.y * 32;
    const int b     = blockIdx.z;

    const float* __restrict__ f1B = f1 + (size_t)b * MIDc * HW;
    const float* __restrict__ f2B = f2 + (size_t)b * MIDc * HW;

    v8f c[2][2] = {};
#pragma unroll 2
    for (int k0 = 0; k0 < MIDc; k0 += 4) {
        const int kk = k0 + 2 * half;
        v2f a[2];   // A = f1^T : A[i][k] = f1[k][i]
        v2f bb[2];  // B = f2   : B[k][j]
#pragma unroll
        for (int t = 0; t < 2; ++t) {
            a[t].x  = f1B[(size_t)kk * HW + iBase + 16 * t + l];
            a[t].y  = f1B[(size_t)(kk + 1) * HW + iBase + 16 * t + l];
            bb[t].x = f2B[(size_t)kk * HW + jBase + 16 * t + l];
            bb[t].y = f2B[(size_t)(kk + 1) * HW + jBase + 16 * t + l];
        }
#pragma unroll
        for (int ii = 0; ii < 2; ++ii)
#pragma unroll
            for (int jj = 0; jj < 2; ++jj)
                c[ii][jj] = WMMA_F32_16x16x4(a[ii], bb[jj], c[ii][jj]);
    }

    float* __restrict__ o = attn + (size_t)b * HW * HW
                          + (size_t)(iBase + 8 * half) * HW + jBase + l;
#pragma unroll
    for (int ii = 0; ii < 2; ++ii)
#pragma unroll
        for (int jj = 0; jj < 2; ++jj)
#pragma unroll
            for (int r = 0; r < 8; ++r)
                o[(size_t)(16 * ii + r) * HW + 16 * jj] = c[ii][jj][r];
}

// ---------------------------------------------------------------------------
// Kernel 3: in-place row softmax over attn rows (B*N rows of length N=576).
//   One wave32 per row: 18 elems/lane, shfl_xor (ds-permute) tree reductions.
// ---------------------------------------------------------------------------
__global__ __launch_bounds__(256)
void k_softmax_rows(float* __restrict__ attn)
{
    const int lane = threadIdx.x & 31;
    const int wave = threadIdx.x >> 5;                  // 8 waves / block
    const size_t row = (size_t)blockIdx.x * 8 + wave;   // < Bsz*HW
    float* __restrict__ p = attn + row * HW;

    float vals[HW / 32];
    float m = -3.402823466e38f;
#pragma unroll
    for (int t = 0; t < HW / 32; ++t) {
        vals[t] = p[lane + t * 32];
        m = fmaxf(m, vals[t]);
    }
#pragma unroll
    for (int off = 16; off > 0; off >>= 1)
        m = fmaxf(m, __shfl_xor(m, off, 32));

    float s = 0.0f;
#pragma unroll
    for (int t = 0; t < HW / 32; ++t) {
        vals[t] = __expf(vals[t] - m);
        s += vals[t];
    }
#pragma unroll
    for (int off = 16; off > 0; off >>= 1)
        s += __shfl_xor(s, off, 32);

    const float inv = 1.0f / s;
#pragma unroll
    for (int t = 0; t < HW / 32; ++t)
        p[lane + t * 32] = vals[t] * inv;
}

// ---------------------------------------------------------------------------
// Kernel 4: out[b][c][i] = alpha * sum_j attn[b][i][j]*cam[b][c][j] + cam[b][c][i]
//   GEMM per batch: A = cam (C x N over j), B = attn^T (j x i), K = N(576).
//   Both fragment types are contiguous b64 loads. Register blocking:
//   2 channel-tiles x 4 spatial-tiles per wave (32 x 64 output block),
//   8 WMMAs per 6 b64 loads; cuts cam L2 re-reads 4x, attn re-reads 2x.
// ---------------------------------------------------------------------------
__global__ __launch_bounds__(32)
void k_apply_attn(const float* __restrict__ cam,
                  const float* __restrict__ attn,
                  const float* __restrict__ alphaPtr,
                  float* __restrict__ out)
{
    const int lane = threadIdx.x & 31;
    const int half = lane >> 4;
    const int l    = lane & 15;
    const int nBase = blockIdx.x * 64;  // output spatial block (i): 4 tiles
    const int cBase = blockIdx.y * 32;  // channel block: 2 tiles
    const int b     = blockIdx.z;

    const float* __restrict__ camB  = cam + (size_t)b * Cch * HW;
    const float* __restrict__ attnB = attn + (size_t)b * HW * HW;

    const float* __restrict__ camRow0 = camB + (size_t)(cBase + l) * HW;       // A rows (c)
    const float* __restrict__ camRow1 = camRow0 + (size_t)16 * HW;
    const float* __restrict__ pRow[4] = {                                      // B "cols" (i)
        attnB + (size_t)(nBase + 0  + l) * HW,
        attnB + (size_t)(nBase + 16 + l) * HW,
        attnB + (size_t)(nBase + 32 + l) * HW,
        attnB + (size_t)(nBase + 48 + l) * HW,
    };

    v8f acc[2][4] = {};
#pragma unroll 2
    for (int k0 = 0; k0 < HW; k0 += 4) {
        const int kk = k0 + 2 * half;
        v2f a[2];
        a[0] = *(const v2f*)(camRow0 + kk);   // cam[b][c][j..j+1]
        a[1] = *(const v2f*)(camRow1 + kk);
        v2f bb[4];
#pragma unroll
        for (int ni = 0; ni < 4; ++ni)
            bb[ni] = *(const v2f*)(pRow[ni] + kk);  // attn[b][i][j..j+1]
#pragma unroll
        for (int mi = 0; mi < 2; ++mi)
#pragma unroll
            for (int ni = 0; ni < 4; ++ni)
                acc[mi][ni] = WMMA_F32_16x16x4(a[mi], bb[ni], acc[mi][ni]);
    }

    const float alpha = alphaPtr[0];
    const float* __restrict__ ci = camB + (size_t)(cBase + 8 * half) * HW + nBase + l;
    float* __restrict__ o = out + ((size_t)b * Cch + cBase + 8 * half) * HW + nBase + l;
#pragma unroll
    for (int mi = 0; mi < 2; ++mi)
#pragma unroll
        for (int ni = 0; ni < 4; ++ni)
#pragma unroll
            for (int r = 0; r < 8; ++r) {
                const size_t off = (size_t)(16 * mi + r) * HW + 16 * ni;
                o[off] = alpha * acc[mi][ni][r] + ci[off];
            }
}

// ---------------------------------------------------------------------------
// Host-side launch. Workspace layout (floats):
//   f1   : B*MID*N            (18.9 MB)
//   f2   : B*MID*N            (18.9 MB)
//   attn : B*N*N              (42.5 MB)
// ---------------------------------------------------------------------------
extern "C" void kernel_launch(void* const* d_in, const int* in_sizes, int n_in,
                              void* d_out, int out_size, void* d_ws, size_t ws_size,
                              hipStream_t stream)
{
    (void)in_sizes; (void)n_in; (void)out_size; (void)ws_size;

    const float* cam   = (const float*)d_in[0];
    const float* feat  = (const float*)d_in[1];
    const float* W1    = (const float*)d_in[2];
    const float* W2    = (const float*)d_in[3];
    const float* alpha = (const float*)d_in[4];
    float* out = (float*)d_out;

    float* f1   = (float*)d_ws;
    float* f2   = f1 + (size_t)Bsz * MIDc * HW;
    float* attn = f2 + (size_t)Bsz * MIDc * HW;

    // Stage 1: f1/f2 projections (shared feat stream), 32x32 block per wave
    k_conv1x1_pair<<<dim3(HW / 32, MIDc / 32, Bsz), 32, 0, stream>>>(feat, W1, W2, f1, f2);

    // Stage 2: attention scores, 32x32 block per wave
    k_attn_scores<<<dim3(HW / 32, HW / 32, Bsz), 32, 0, stream>>>(f1, f2, attn);

    // Stage 3: row softmax (in place), 8 rows per 256-thread block
    k_softmax_rows<<<dim3((Bsz * HW) / 8), 256, 0, stream>>>(attn);

    // Stage 4: refine cam with attention + residual, 32x64 block per wave
    k_apply_attn<<<dim3(HW / 64, Cch / 32, Bsz), 32, 0, stream>>>(cam, attn, alpha, out);
}